// sparse_conv_20907900796985
// MI455X (gfx1250) — compile-verified
//
#include <hip/hip_runtime.h>
#include <hip/hip_bf16.h>

typedef float v2f __attribute__((ext_vector_type(2)));
typedef float v8f __attribute__((ext_vector_type(8)));

// Fixed geometry from reference setup: N=8, Cin=64, H=W=256.
#define DH 256
#define DW 256
#define DHW 65536        // 2^16
#define DN 8
#define DC 64

// f32 WMMA fragments (wave32):
//  A (16x4):  lane L -> M = L&15 ; K = ((L>=16)?2:0) + vgpr{0,1}
//  B (4x16):  lane L -> N = L&15 ; K = ((L>=16)?2:0) + vgpr{0,1}
//  C/D:       vgpr i -> M = ((L>=16)?8:0) + i ; N = L&15
static __device__ __forceinline__ v8f wmma4(v2f a, v2f b, v8f c) {
    return __builtin_amdgcn_wmma_f32_16x16x4_f32(false, a, false, b,
                                                 (short)0, c, false, false);
}
#define V8F_ZERO {0.f,0.f,0.f,0.f,0.f,0.f,0.f,0.f}

// ---------------------------------------------------------------------------
// 1. Dense 1x1 conv + bias + ReLU (pre-BN).  One wave = 16 rows x 64 channels
//    (4 accumulators) so the A fragment is loaded exactly once per wave.
//    Block = 8 waves = 128 rows.
// ---------------------------------------------------------------------------
__global__ __launch_bounds__(256) void dense_conv1x1_kernel(
    const float* __restrict__ x, const float* __restrict__ Wc,
    const float* __restrict__ bc, float* __restrict__ y)
{
    const int Cin = DC, Cout = DC;
    int lane  = threadIdx.x & 31;
    int wave  = threadIdx.x >> 5;
    int m0    = (blockIdx.x * 8 + wave) * 16;
    int laneN = lane & 15;
    int kaOff = (lane >> 4) << 1;

    int mA  = m0 + laneN;
    int nA  = mA >> 16;
    int hwA = mA & (DHW - 1);
    const float* xb = x + (size_t)nA * Cin * DHW + hwA;
    const float* wr = Wc + (size_t)laneN * Cin;     // + j*16*Cin for col tile j

    v8f acc[4] = {V8F_ZERO, V8F_ZERO, V8F_ZERO, V8F_ZERO};
    #pragma unroll
    for (int k0 = 0; k0 < Cin; k0 += 4) {
        int ka = k0 + kaOff;
        v2f a;
        a.x = xb[(size_t)ka * DHW];
        a.y = xb[(size_t)(ka + 1) * DHW];
        #pragma unroll
        for (int j = 0; j < 4; ++j) {
            v2f b;
            b.x = wr[(size_t)j * 16 * Cin + ka];
            b.y = wr[(size_t)j * 16 * Cin + ka + 1];
            acc[j] = wmma4(a, b, acc[j]);
        }
    }
    int mhalf = (lane >> 4) * 8;
    #pragma unroll
    for (int j = 0; j < 4; ++j) {
        int o = j * 16 + laneN;
        float bias = bc[o];
        #pragma unroll
        for (int i = 0; i < 8; ++i) {
            int mi  = m0 + mhalf + i;
            int ni  = mi >> 16;
            int hwi = mi & (DHW - 1);
            float v = fmaxf(acc[j][i] + bias, 0.f);
            y[(size_t)ni * Cout * DHW + (size_t)o * DHW + hwi] = v;
        }
    }
}

// ---------------------------------------------------------------------------
// 2. Per-channel sum / sumsq over an (Nplanes, C, P) buffer, P = 2^logP
// ---------------------------------------------------------------------------
__global__ __launch_bounds__(256) void channel_stats_kernel(
    const float* __restrict__ buf, float* __restrict__ outSum,
    float* __restrict__ outSq, int C, int logP, int nPlanes)
{
    int c = blockIdx.x;
    long T = (long)nPlanes << logP;
    long stride = (long)gridDim.y * blockDim.x;
    long pm = (1L << logP) - 1;
    float s = 0.f, q = 0.f;
    for (long j = (long)blockIdx.y * blockDim.x + threadIdx.x; j < T; j += stride) {
        long n = j >> logP;
        long p = j & pm;
        float v = buf[((n * C + c) << logP) + p];
        s += v; q += v * v;
    }
    __shared__ float ls[256], lq[256];
    int tid = threadIdx.x;
    ls[tid] = s; lq[tid] = q;
    __syncthreads();
    for (int off = 128; off > 0; off >>= 1) {
        if (tid < off) { ls[tid] += ls[tid + off]; lq[tid] += lq[tid + off]; }
        __syncthreads();
    }
    if (tid == 0) { atomicAdd(&outSum[c], ls[0]); atomicAdd(&outSq[c], lq[0]); }
}

// ---------------------------------------------------------------------------
// 3. BN fold: scale = g*rsqrt(var+eps), shift = beta - mean*scale
// ---------------------------------------------------------------------------
__global__ void finalize_bn_kernel(const float* __restrict__ sum,
                                   const float* __restrict__ sq,
                                   const float* __restrict__ gamma,
                                   const float* __restrict__ beta,
                                   float* __restrict__ scale,
                                   float* __restrict__ shift, int C, float invT)
{
    int c = blockIdx.x * blockDim.x + threadIdx.x;
    if (c >= C) return;
    float m  = sum[c] * invT;
    float v  = sq[c] * invT - m * m;
    float rs = rsqrtf(v + 1e-5f);
    float s  = gamma[c] * rs;
    scale[c] = s;
    shift[c] = beta[c] - m * s;
}

// ---------------------------------------------------------------------------
// 4. In-place per-channel affine on the dense output (NCHW, C=64, HW=2^16)
// ---------------------------------------------------------------------------
__global__ __launch_bounds__(256) void affine_dense_kernel(
    float* __restrict__ y, const float* __restrict__ sc,
    const float* __restrict__ sh, int total4)
{
    int i = blockIdx.x * blockDim.x + threadIdx.x;
    if (i >= total4) return;
    int c = ((i << 2) >> 16) & (DC - 1);
    float4* p = reinterpret_cast<float4*>(y) + i;
    float4 v = *p;
    float s = sc[c], t = sh[c];
    v.x = v.x * s + t; v.y = v.y * s + t; v.z = v.z * s + t; v.w = v.w * s + t;
    *p = v;
}

// ---------------------------------------------------------------------------
// 5. Gather + 1x1 conv (64 -> 128) + bias + ReLU (pre-BN).
//    One wave = 16 rows x 128 channels (8 accumulators); gathered A fragment
//    is loaded from the normalized dense output exactly once.
// ---------------------------------------------------------------------------
__global__ __launch_bounds__(256) void gather_conv1_kernel(
    const float* __restrict__ xn, const int* __restrict__ abi,
    const float* __restrict__ W1, const float* __restrict__ b1w,
    float* __restrict__ gout, const int* __restrict__ pbs,
    const int* __restrict__ pbo, const int* __restrict__ pbt)
{
    const int Cin = 64, Cout = 128;
    int bs = pbs[0], boff = pbo[0], bstr = pbt[0];
    int PP = bs * bs;                       // 64
    int lane  = threadIdx.x & 31;
    int wave  = threadIdx.x >> 5;
    int rowbase = (blockIdx.x * 8 + wave) * 16;
    int blk  = rowbase / PP;
    int pix0 = rowbase - blk * PP;
    int laneN = lane & 15;
    int pixA = pix0 + laneN;
    int py = pixA / bs, px = pixA - py * bs;

    int b_ = abi[3 * blk], by = abi[3 * blk + 1], bx = abi[3 * blk + 2];
    int row = boff + by * bstr + py;
    int col = boff + bx * bstr + px;
    const float* xb = xn + (size_t)b_ * Cin * DHW + row * DW + col;

    int kaOff = (lane >> 4) << 1;
    const float* wr = W1 + (size_t)laneN * Cin;

    v8f acc[8] = {V8F_ZERO, V8F_ZERO, V8F_ZERO, V8F_ZERO,
                  V8F_ZERO, V8F_ZERO, V8F_ZERO, V8F_ZERO};
    #pragma unroll
    for (int k0 = 0; k0 < Cin; k0 += 4) {
        int ka = k0 + kaOff;
        v2f a;
        a.x = xb[(size_t)ka * DHW];
        a.y = xb[(size_t)(ka + 1) * DHW];
        #pragma unroll
        for (int j = 0; j < 8; ++j) {
            v2f b;
            b.x = wr[(size_t)j * 16 * Cin + ka];
            b.y = wr[(size_t)j * 16 * Cin + ka + 1];
            acc[j] = wmma4(a, b, acc[j]);
        }
    }
    int mhalf = (lane >> 4) * 8;
    #pragma unroll
    for (int j = 0; j < 8; ++j) {
        int o = j * 16 + laneN;
        float bias = b1w[o];
        #pragma unroll
        for (int i = 0; i < 8; ++i) {
            int pixD = pix0 + mhalf + i;
            float v = fmaxf(acc[j][i] + bias, 0.f);
            gout[(size_t)blk * Cout * PP + (size_t)o * PP + pixD] = v;
        }
    }
}

// ---------------------------------------------------------------------------
// 6. 3x3 SAME conv inside each 8x8 tile (128 -> 128).  One workgroup per
//    block: the 128x64 input tile (32 KB) is staged in LDS with the stage-1
//    BN affine pre-applied (padding is post-BN zero, as in the reference).
//    Wave w: rows = (w&3)*16..+15, out channels = (w>>2)*64..+63 (4 accs).
//    Implicit im2col, K ordered tap-major x channel.
// ---------------------------------------------------------------------------
__global__ __launch_bounds__(256) void conv3x3_kernel(
    const float* __restrict__ gin, const float* __restrict__ sc,
    const float* __restrict__ sh, const float* __restrict__ W2,
    const float* __restrict__ b2w, float* __restrict__ gout,
    const int* __restrict__ pbs)
{
    const int Cin = 128, Cout = 128;
    __shared__ float tile[Cin * 64];
    int bs = pbs[0];
    int PP = bs * bs;
    int blk = blockIdx.x;
    const float* gblk = gin + (size_t)blk * Cin * PP;

    for (int idx = threadIdx.x; idx < Cin * PP; idx += 256) {
        int c = idx / PP;
        tile[idx] = gblk[idx] * sc[c] + sh[c];
    }
    __syncthreads();

    int lane  = threadIdx.x & 31;
    int wave  = threadIdx.x >> 5;
    int pix0  = (wave & 3) * 16;
    int oBase = (wave >> 2) * 64;
    int laneN = lane & 15;
    int pixA  = pix0 + laneN;
    int py = pixA / bs, px = pixA - py * bs;
    int kaOff = (lane >> 4) << 1;
    const float* wrow = W2 + (size_t)(oBase + laneN) * Cin * 9;

    v8f acc[4] = {V8F_ZERO, V8F_ZERO, V8F_ZERO, V8F_ZERO};
    for (int t = 0; t < 9; ++t) {
        int dy = t / 3, dx = t - dy * 3;
        int yy = py + dy - 1, xx = px + dx - 1;
        bool valid = (yy >= 0) && (yy < bs) && (xx >= 0) && (xx < bs);
        const float* lp = tile + yy * bs + xx;     // + c*PP (LDS)
        const float* wp = wrow + dy * 3 + dx;      // + c*9
        #pragma unroll 4
        for (int c0 = 0; c0 < Cin; c0 += 4) {
            int ca = c0 + kaOff;
            v2f a;
            a.x = valid ? lp[(size_t)ca * PP] : 0.f;
            a.y = valid ? lp[(size_t)(ca + 1) * PP] : 0.f;
            #pragma unroll
            for (int j = 0; j < 4; ++j) {
                v2f b;
                b.x = wp[((size_t)j * 16 * Cin + ca) * 9];
                b.y = wp[((size_t)j * 16 * Cin + ca + 1) * 9];
                acc[j] = wmma4(a, b, acc[j]);
            }
        }
    }
    int mhalf = (lane >> 4) * 8;
    #pragma unroll
    for (int j = 0; j < 4; ++j) {
        int o = oBase + j * 16 + laneN;
        float bias = b2w[o];
        #pragma unroll
        for (int i = 0; i < 8; ++i) {
            int pixD = pix0 + mhalf + i;
            float v = fmaxf(acc[j][i] + bias, 0.f);
            gout[(size_t)blk * Cout * PP + (size_t)o * PP + pixD] = v;
        }
    }
}

// ---------------------------------------------------------------------------
// 7. 1x1 conv (128 -> 64) with fused stage-2 BN affine on the input.
//    One wave = 16 rows x 64 channels (4 accumulators).
// ---------------------------------------------------------------------------
__global__ __launch_bounds__(256) void conv1x1_g3_kernel(
    const float* __restrict__ gin, const float* __restrict__ sc,
    const float* __restrict__ sh, const float* __restrict__ W3,
    const float* __restrict__ b3w, float* __restrict__ gout,
    const int* __restrict__ pbs)
{
    const int Cin = 128, Cout = 64;
    int bs = pbs[0];
    int PP = bs * bs;
    int lane  = threadIdx.x & 31;
    int wave  = threadIdx.x >> 5;
    int rowbase = (blockIdx.x * 8 + wave) * 16;
    int blk  = rowbase / PP;
    int pix0 = rowbase - blk * PP;
    int laneN = lane & 15;
    int pixA = pix0 + laneN;
    int kaOff = (lane >> 4) << 1;

    const float* gb = gin + (size_t)blk * Cin * PP + pixA;
    const float* wr = W3 + (size_t)laneN * Cin;

    v8f acc[4] = {V8F_ZERO, V8F_ZERO, V8F_ZERO, V8F_ZERO};
    #pragma unroll 8
    for (int k0 = 0; k0 < Cin; k0 += 4) {
        int ka = k0 + kaOff;
        v2f a;
        a.x = gb[(size_t)ka * PP] * sc[ka] + sh[ka];
        a.y = gb[(size_t)(ka + 1) * PP] * sc[ka + 1] + sh[ka + 1];
        #pragma unroll
        for (int j = 0; j < 4; ++j) {
            v2f b;
            b.x = wr[(size_t)j * 16 * Cin + ka];
            b.y = wr[(size_t)j * 16 * Cin + ka + 1];
            acc[j] = wmma4(a, b, acc[j]);
        }
    }
    int mhalf = (lane >> 4) * 8;
    #pragma unroll
    for (int j = 0; j < 4; ++j) {
        int o = j * 16 + laneN;
        float bias = b3w[o];
        #pragma unroll
        for (int i = 0; i < 8; ++i) {
            int pixD = pix0 + mhalf + i;
            float v = fmaxf(acc[j][i] + bias, 0.f);
            gout[(size_t)blk * Cout * PP + (size_t)o * PP + pixD] = v;
        }
    }
}

// ---------------------------------------------------------------------------
// 8. Scatter-add with fused stage-3 BN affine (atomics for duplicate blocks)
// ---------------------------------------------------------------------------
__global__ __launch_bounds__(256) void scatter_add_kernel(
    const float* __restrict__ g, const float* __restrict__ sc,
    const float* __restrict__ sh, const int* __restrict__ abi,
    float* __restrict__ y, const int* __restrict__ pbs,
    const int* __restrict__ pbo, const int* __restrict__ pbt, int nB)
{
    const int C = 64;
    int bs = pbs[0], boff = pbo[0], bstr = pbt[0];
    int PP = bs * bs;
    long total = (long)nB * C * PP;
    long i = (long)blockIdx.x * blockDim.x + threadIdx.x;
    if (i >= total) return;
    int pix = (int)(i % PP);
    long r = i / PP;
    int c = (int)(r % C);
    int blk = (int)(r / C);
    int py = pix / bs, px = pix - py * bs;
    int b_ = abi[3 * blk], by = abi[3 * blk + 1], bx = abi[3 * blk + 2];
    int row = boff + by * bstr + py;
    int col = boff + bx * bstr + px;
    float v = g[i] * sc[c] + sh[c];
    atomicAdd(&y[((size_t)b_ * C + c) * DHW + row * DW + col], v);
}

// ---------------------------------------------------------------------------
extern "C" void kernel_launch(void* const* d_in, const int* in_sizes, int n_in,
                              void* d_out, int out_size, void* d_ws, size_t ws_size,
                              hipStream_t stream)
{
    const float* x   = (const float*)d_in[0];
    const int*   abi = (const int*)  d_in[1];
    const int*   pbs = (const int*)  d_in[2];   // bsize
    const int*   pbo = (const int*)  d_in[4];   // boffset
    const int*   pbt = (const int*)  d_in[5];   // bstride
    const float* Wc  = (const float*)d_in[6];
    const float* bc  = (const float*)d_in[7];
    const float* gc  = (const float*)d_in[8];
    const float* btc = (const float*)d_in[9];
    const float* W1  = (const float*)d_in[10];
    const float* b1  = (const float*)d_in[11];
    const float* g1  = (const float*)d_in[12];
    const float* bt1 = (const float*)d_in[13];
    const float* W2  = (const float*)d_in[14];
    const float* b2  = (const float*)d_in[15];
    const float* g2  = (const float*)d_in[16];
    const float* bt2 = (const float*)d_in[17];
    const float* W3  = (const float*)d_in[18];
    const float* b3  = (const float*)d_in[19];
    const float* g3g = (const float*)d_in[20];
    const float* bt3 = (const float*)d_in[21];

    float* out = (float*)d_out;
    float* ws  = (float*)d_ws;
    int nB = in_sizes[1] / 3;                   // 4096
    const int PP = 64;                          // bsize^2 (host tiling constant)

    float* bufA = ws;                           // nB*128*64 f32 (also stage-3 out)
    float* bufB = ws + (size_t)nB * 128 * PP;
    float* st   = ws + 2 * (size_t)nB * 128 * PP;
    float *sum0 = st,        *sq0 = st + 64,   *sc0 = st + 128,  *sh0 = st + 192;
    float *sum1 = st + 256,  *sq1 = st + 384,  *sc1 = st + 512,  *sh1 = st + 640;
    float *sum2 = st + 768,  *sq2 = st + 896,  *sc2 = st + 1024, *sh2 = st + 1152;
    float *sum3 = st + 1280, *sq3 = st + 1344, *sc3 = st + 1408, *sh3 = st + 1472;

    hipMemsetAsync(st, 0, 1536 * sizeof(float), stream);

    // Dense: conv1x1 + relu -> out (pre-BN), stats, fold, affine in place
    dense_conv1x1_kernel<<<(DN * DHW) / 128, 256, 0, stream>>>(x, Wc, bc, out);
    channel_stats_kernel<<<dim3(64, 64), 256, 0, stream>>>(out, sum0, sq0, 64, 16, DN);
    finalize_bn_kernel<<<1, 64, 0, stream>>>(sum0, sq0, gc, btc, sc0, sh0, 64,
                                             1.f / (float)(DN * DHW));
    affine_dense_kernel<<<(DN * DC * DHW / 4 + 255) / 256, 256, 0, stream>>>(
        out, sc0, sh0, DN * DC * DHW / 4);

    int rowBlocks = nB * PP / 128;              // 2048 (8 waves x 16 rows each)

    // Stage 1: gather + 1x1 (64->128)
    gather_conv1_kernel<<<rowBlocks, 256, 0, stream>>>(out, abi, W1, b1, bufA,
                                                       pbs, pbo, pbt);
    channel_stats_kernel<<<dim3(128, 32), 256, 0, stream>>>(bufA, sum1, sq1, 128, 6, nB);
    finalize_bn_kernel<<<1, 128, 0, stream>>>(sum1, sq1, g1, bt1, sc1, sh1, 128,
                                              1.f / (float)(nB * PP));

    // Stage 2: 3x3 (128->128), stage-1 BN fused into the LDS-staged tile
    conv3x3_kernel<<<nB, 256, 0, stream>>>(bufA, sc1, sh1, W2, b2, bufB, pbs);
    channel_stats_kernel<<<dim3(128, 32), 256, 0, stream>>>(bufB, sum2, sq2, 128, 6, nB);
    finalize_bn_kernel<<<1, 128, 0, stream>>>(sum2, sq2, g2, bt2, sc2, sh2, 128,
                                              1.f / (float)(nB * PP));

    // Stage 3: 1x1 (128->64), stage-2 BN fused on input; output reuses bufA
    conv1x1_g3_kernel<<<rowBlocks, 256, 0, stream>>>(bufB, sc2, sh2, W3, b3, bufA, pbs);
    channel_stats_kernel<<<dim3(64, 32), 256, 0, stream>>>(bufA, sum3, sq3, 64, 6, nB);
    finalize_bn_kernel<<<1, 64, 0, stream>>>(sum3, sq3, g3g, bt3, sc3, sh3, 64,
                                             1.f / (float)(nB * PP));

    // Scatter-add with stage-3 BN fused
    long total = (long)nB * 64 * PP;
    scatter_add_kernel<<<(int)((total + 255) / 256), 256, 0, stream>>>(
        bufA, sc3, sh3, abi, out, pbs, pbo, pbt, nB);
}